// S4Block_1855425872534
// MI455X (gfx1250) — compile-verified
//
#include <hip/hip_runtime.h>

// ---------------------------------------------------------------------------
// S4 block for MI455X (gfx1250, wave32):
//   out = x + ( SSM( LN(x) @ W_in^T + b_in ) @ W_out^T + b_out )
// Shapes: B=2, S=2048, d=768, N=64.
// ---------------------------------------------------------------------------

#define D_MODEL 768
#define D_STATE 64
#define BATCH   2
#define SEQ     2048

typedef __attribute__((ext_vector_type(2))) float v2f;
typedef __attribute__((ext_vector_type(8))) float v8f;

// ---------------------------------------------------------------------------
// Kernel 1: LayerNorm, one 256-thread block per row of 768.
// ---------------------------------------------------------------------------
__global__ __launch_bounds__(256) void ln_kernel(const float* __restrict__ x,
                                                 const float* __restrict__ gamma,
                                                 const float* __restrict__ beta,
                                                 float* __restrict__ out) {
    __shared__ float red[8];
    const int row  = blockIdx.x;
    const int tid  = threadIdx.x;
    const int lane = tid & 31;
    const int wave = tid >> 5;
    const float* xr = x + (size_t)row * D_MODEL;

    float v0 = xr[tid], v1 = xr[tid + 256], v2 = xr[tid + 512];

    // mean
    float s = v0 + v1 + v2;
    #pragma unroll
    for (int off = 16; off > 0; off >>= 1) s += __shfl_xor(s, off, 32);
    if (lane == 0) red[wave] = s;
    __syncthreads();
    float tot = 0.f;
    #pragma unroll
    for (int i = 0; i < 8; ++i) tot += red[i];
    const float mean = tot * (1.0f / D_MODEL);
    __syncthreads();

    // variance
    float d0 = v0 - mean, d1 = v1 - mean, d2 = v2 - mean;
    float q = d0 * d0 + d1 * d1 + d2 * d2;
    #pragma unroll
    for (int off = 16; off > 0; off >>= 1) q += __shfl_xor(q, off, 32);
    if (lane == 0) red[wave] = q;
    __syncthreads();
    float qtot = 0.f;
    #pragma unroll
    for (int i = 0; i < 8; ++i) qtot += red[i];
    const float rstd = rsqrtf(qtot * (1.0f / D_MODEL) + 1e-5f);

    float* orow = out + (size_t)row * D_MODEL;
    orow[tid]       = d0 * rstd * gamma[tid]       + beta[tid];
    orow[tid + 256] = d1 * rstd * gamma[tid + 256] + beta[tid + 256];
    orow[tid + 512] = d2 * rstd * gamma[tid + 512] + beta[tid + 512];
}

// ---------------------------------------------------------------------------
// Kernel 2/4: fp32 WMMA GEMM  Out[M,N] = A[M,K] @ W[N,K]^T + bias (+ resid).
// Uses V_WMMA_F32_16X16X4_F32. Block = 8 waves; waves arranged 4(M) x 2(N),
// each wave owns one 16x16 output tile -> block tile 64x32.
//
// A-frag (16x4 f32, 2 VGPRs/lane): lanes 0-15 hold K={k0,k0+1} of row m0+l,
// lanes 16-31 hold K={k0+2,k0+3}. B-frag mirrors with N across lanes; since
// Bmat[k][n] = W[n][k], both frags are contiguous float2 loads.
// C/D (16x16 f32, 8 VGPRs): VGPR r = row m0 + 8*half + r, col n0 + l.
// ---------------------------------------------------------------------------
__global__ __launch_bounds__(256) void gemm_wmma_f32(const float* __restrict__ A,
                                                     const float* __restrict__ W,
                                                     const float* __restrict__ bias,
                                                     const float* __restrict__ resid,
                                                     float* __restrict__ Out,
                                                     int M, int N, int K) {
    const int tid  = threadIdx.x;
    const int lane = tid & 31;
    const int wave = tid >> 5;
    const int wm   = wave & 3;      // 0..3 along M
    const int wn   = wave >> 2;     // 0..1 along N
    const int m0   = blockIdx.x * 64 + wm * 16;
    const int n0   = blockIdx.y * 32 + wn * 16;
    const int half = lane >> 4;     // 0: lanes 0-15, 1: lanes 16-31
    const int l    = lane & 15;

    const float* arow = A + (size_t)(m0 + l) * K + 2 * half;
    const float* brow = W + (size_t)(n0 + l) * K + 2 * half;

    v8f c = {};
    #pragma unroll 4
    for (int k0 = 0; k0 < K; k0 += 4) {
        v2f a = *(const v2f*)(arow + k0);
        v2f b = *(const v2f*)(brow + k0);
        c = __builtin_amdgcn_wmma_f32_16x16x4_f32(
                /*neg_a=*/false, a, /*neg_b=*/false, b,
                /*c_mod=*/(short)0, c, /*reuse_a=*/false, /*reuse_b=*/false);
    }

    const int n   = n0 + l;
    const float bv = bias[n];
    #pragma unroll
    for (int r = 0; r < 8; ++r) {
        const int m = m0 + half * 8 + r;
        float v = c[r] + bv;
        if (resid != nullptr) v += resid[(size_t)m * N + n];
        Out[(size_t)m * N + n] = v;
    }
}

// ---------------------------------------------------------------------------
// Kernel 3: diagonal SSM scan. One wave32 per (batch, channel); each lane
// owns states n = {2*lane, 2*lane+1}. Per step: 2 FMAs for h, butterfly
// reduce of h.C over 32 lanes, lane 0 stores y.
// ---------------------------------------------------------------------------
__global__ __launch_bounds__(256) void s4_scan(const float* __restrict__ u,
                                               const float* __restrict__ log_A,
                                               const float* __restrict__ B_p,
                                               const float* __restrict__ C_p,
                                               const float* __restrict__ D_p,
                                               const float* __restrict__ log_dt,
                                               float* __restrict__ y) {
    const int lane = threadIdx.x & 31;
    const int wave = threadIdx.x >> 5;
    const int ch   = blockIdx.x * 8 + wave;     // 0 .. BATCH*D_MODEL-1
    const int b    = ch / D_MODEL;
    const int d    = ch % D_MODEL;

    const float dt = __expf(log_dt[d]);
    const v2f la = ((const v2f*)(log_A + (size_t)d * D_STATE))[lane];
    v2f Ab;
    Ab.x = __expf(dt * (-__expf(la.x)));
    Ab.y = __expf(dt * (-__expf(la.y)));
    v2f Bb = ((const v2f*)(B_p + (size_t)d * D_STATE))[lane];
    Bb.x *= dt; Bb.y *= dt;
    const v2f Cc = ((const v2f*)(C_p + (size_t)d * D_STATE))[lane];
    const float Dv = D_p[d];

    const float* ub = u + (size_t)b * SEQ * D_MODEL + d;
    float*       yb = y + (size_t)b * SEQ * D_MODEL + d;

    v2f h = {0.f, 0.f};
    for (int s = 0; s < SEQ; ++s) {
        const float ut = ub[(size_t)s * D_MODEL];
        h.x = Ab.x * h.x + Bb.x * ut;
        h.y = Ab.y * h.y + Bb.y * ut;
        float p = h.x * Cc.x + h.y * Cc.y;
        #pragma unroll
        for (int off = 16; off > 0; off >>= 1) p += __shfl_xor(p, off, 32);
        if (lane == 0) yb[(size_t)s * D_MODEL] = p + Dv * ut;
    }
}

// ---------------------------------------------------------------------------
// Launch
// ---------------------------------------------------------------------------
extern "C" void kernel_launch(void* const* d_in, const int* in_sizes, int n_in,
                              void* d_out, int out_size, void* d_ws, size_t ws_size,
                              hipStream_t stream) {
    const float* x        = (const float*)d_in[0];
    const float* ln_gamma = (const float*)d_in[1];
    const float* ln_beta  = (const float*)d_in[2];
    const float* W_in     = (const float*)d_in[3];
    const float* b_in     = (const float*)d_in[4];
    const float* log_A    = (const float*)d_in[5];
    const float* B_p      = (const float*)d_in[6];
    const float* C_p      = (const float*)d_in[7];
    const float* D_p      = (const float*)d_in[8];
    const float* log_dt   = (const float*)d_in[9];
    const float* W_out    = (const float*)d_in[10];
    const float* b_out    = (const float*)d_in[11];
    float* out = (float*)d_out;

    const int M = BATCH * SEQ;                 // 4096 rows
    const size_t plane = (size_t)M * D_MODEL;  // 3,145,728 floats

    float* xn = (float*)d_ws;                  // LN output
    float* u  = xn + plane;                    // in_proj output
    float* yv = u + plane;                     // SSM output

    // 1) LayerNorm
    ln_kernel<<<M, 256, 0, stream>>>(x, ln_gamma, ln_beta, xn);

    // 2) u = xn @ W_in^T + b_in
    dim3 gg(M / 64, D_MODEL / 32);
    gemm_wmma_f32<<<gg, 256, 0, stream>>>(xn, W_in, b_in, nullptr, u,
                                          M, D_MODEL, D_MODEL);

    // 3) SSM scan (wave per (batch, channel))
    s4_scan<<<(BATCH * D_MODEL) / 8, 256, 0, stream>>>(u, log_A, B_p, C_p,
                                                       D_p, log_dt, yv);

    // 4) out = yv @ W_out^T + b_out + x  (residual fused)
    gemm_wmma_f32<<<gg, 256, 0, stream>>>(yv, W_out, b_out, x, out,
                                          M, D_MODEL, D_MODEL);
}